// MyersonNet_16063177687587
// MI455X (gfx1250) — compile-verified
//
#include <hip/hip_runtime.h>
#include <math.h>

typedef __attribute__((ext_vector_type(2))) float v2f;
typedef __attribute__((ext_vector_type(4))) float v4f;
typedef __attribute__((ext_vector_type(8))) float v8f;

#define NUM_BUYERS 32
#define BATCHN     524288
#define NTILES     (BATCHN / 16)
#define NEG_BIG    (-3.0e38f)

// One wave handles a 16-row batch tile.
// Per WMMA w (w = 0..15): buyers 2w (M=0..7) and 2w+1 (M=8..15), padded to 8
// param rows each (k-major, j padded to 4 with expw=0 / beta=-BIG).
//   D[m,n] = bid[n, 2w + (m>=8)] * expw[buyer,k,j] + beta[buyer,k,j]
// via K=4 contraction: A[m,:] = { m<8?e:0, m<8?0:e, beta, 0 },
//                      B[:,n] = { bid[n,2w], bid[n,2w+1], 1, 0 }.
// C/D layout: lane<16 -> (N=lane, M=vgpr), lane>=16 -> (N=lane-16, M=vgpr+8),
// so each lane owns one buyer's 8 values for one batch row: min/max is per-lane.
__global__ __launch_bounds__(256) void myerson_wmma_kernel(
    const float* __restrict__ bids,
    const float* __restrict__ w_params,
    const float* __restrict__ beta_params,
    float* __restrict__ out)
{
    __shared__ float expw_pad[256];
    __shared__ float beta_pad[256];

    const int tid = threadIdx.x;
    {   // padded param tables, once per block (blockDim.x == 256)
        const int p = tid;
        const int buyer = p >> 3;
        const int kk = (p >> 2) & 1;
        const int j = p & 3;
        if (j < 3) {
            const int idx = buyer * 6 + kk * 3 + j;
            expw_pad[p] = expf(w_params[idx]);
            beta_pad[p] = beta_params[idx];
        } else {
            expw_pad[p] = 0.0f;     // 0*bid contributes nothing
            beta_pad[p] = NEG_BIG;  // never wins the max_j
        }
    }
    __syncthreads();

    const int lane = tid & 31;
    const int m = lane & 15;
    const bool hi = lane >= 16;

    // A operands (ISA: A 16x4 f32 -> VGPR0 = {K0 | K2}, VGPR1 = {K1 | K3})
    float ax[16], ay[16];
    #pragma unroll
    for (int w = 0; w < 16; ++w) {
        const int p = 16 * w + m;
        if (!hi) {
            const float e = expw_pad[p];
            ax[w] = (m < 8) ? e : 0.0f;  // K=0
            ay[w] = (m < 8) ? 0.0f : e;  // K=1
        } else {
            ax[w] = beta_pad[p];         // K=2
            ay[w] = 0.0f;                // K=3
        }
    }

    const int wavesPerGrid = (gridDim.x * blockDim.x) >> 5;
    const int waveId = (blockIdx.x * blockDim.x + tid) >> 5;

    for (int tile = waveId; tile < NTILES; tile += wavesPerGrid) {
        const int row = tile * 16 + m;

        // bids row (128B) via b128 loads; only lanes 0..15 feed the B matrix
        v4f bid4[8];
        if (!hi) {
            const v4f* rp = (const v4f*)(bids + (size_t)row * NUM_BUYERS);
            #pragma unroll
            for (int i = 0; i < 8; ++i) bid4[i] = rp[i];
        }

        float vv[16];
        #pragma unroll
        for (int w = 0; w < 16; ++w) {
            float bx, by;
            if (!hi) {
                const v4f q = bid4[w >> 1];
                bx = (w & 1) ? q.z : q.x;   // bid[row, 2w]
                by = (w & 1) ? q.w : q.y;   // bid[row, 2w+1]
            } else {
                bx = 1.0f;                  // K=2 row of B (beta multiplier)
                by = 0.0f;                  // K=3 row of B
            }
            v2f a; a.x = ax[w]; a.y = ay[w];
            v2f b; b.x = bx;    b.y = by;
            v8f c = {0.f, 0.f, 0.f, 0.f, 0.f, 0.f, 0.f, 0.f};
            v8f d = __builtin_amdgcn_wmma_f32_16x16x4_f32(
                        false, a, false, b, (short)0, c, false, false);
            // this lane's buyer: d[0..3] = k=0 (j padded), d[4..7] = k=1
            const float h0 = fmaxf(fmaxf(d[0], d[1]), fmaxf(d[2], d[3]));
            const float h1 = fmaxf(fmaxf(d[4], d[5]), fmaxf(d[6], d[7]));
            vv[w] = fminf(h0, h1);
        }

        // local top-2 over this lane's 16 buyers (even buyers in lanes<16,
        // odd in lanes>=16); strict > keeps the lowest index on ties
        float best = NEG_BIG, second = NEG_BIG;
        int bidx = 0;
        #pragma unroll
        for (int w = 0; w < 16; ++w) {
            const float v = vv[w];
            const int buyer = 2 * w + (hi ? 1 : 0);
            const bool nb = v > best;
            second = nb ? best : fmaxf(second, v);
            bidx = nb ? buyer : bidx;
            best = nb ? v : best;
        }

        // merge with the partner half-wave (xor 16)
        const float obest = __shfl_xor(best, 16, 32);
        const float osecond = __shfl_xor(second, 16, 32);
        const int obidx = __shfl_xor(bidx, 16, 32);

        const bool otherWins = (obest > best) || ((obest == best) && (obidx < bidx));
        const int gidx = otherWins ? obidx : bidx;
        const float gsecond = otherWins ? fmaxf(best, osecond)
                                        : fmaxf(obest, second);
        const float pay = fmaxf(gsecond, 0.0f);

        // write one-hot allocations + payments; lane writes 16 cols of its row
        const int cbase = hi ? 16 : 0;
        const size_t rbase = (size_t)row * NUM_BUYERS + cbase;
        v4f* allocP = (v4f*)(out + rbase);
        v4f* payP   = (v4f*)(out + (size_t)BATCHN * NUM_BUYERS + rbase);
        #pragma unroll
        for (int g = 0; g < 4; ++g) {
            v4f a4;
            a4.x = (cbase + 4 * g + 0 == gidx) ? 1.0f : 0.0f;
            a4.y = (cbase + 4 * g + 1 == gidx) ? 1.0f : 0.0f;
            a4.z = (cbase + 4 * g + 2 == gidx) ? 1.0f : 0.0f;
            a4.w = (cbase + 4 * g + 3 == gidx) ? 1.0f : 0.0f;
            v4f p4 = a4 * pay;
            __builtin_nontemporal_store(a4, allocP + g);  // streamed, write-once
            __builtin_nontemporal_store(p4, payP + g);
        }
    }
}

extern "C" void kernel_launch(void* const* d_in, const int* in_sizes, int n_in,
                              void* d_out, int out_size, void* d_ws, size_t ws_size,
                              hipStream_t stream) {
    (void)in_sizes; (void)n_in; (void)out_size; (void)d_ws; (void)ws_size;
    const float* bids = (const float*)d_in[0];
    const float* w    = (const float*)d_in[1];
    const float* beta = (const float*)d_in[2];
    float* out = (float*)d_out;

    // 2048 blocks x 8 waves = 16384 waves; 32768 tiles -> 2 tiles per wave,
    // amortizing the per-wave A-operand setup while saturating the device.
    dim3 block(256);
    dim3 grid(2048);
    myerson_wmma_kernel<<<grid, block, 0, stream>>>(bids, w, beta, out);
}